// MusicHybridEmbedding_56676388438137
// MI455X (gfx1250) — compile-verified
//
#include <hip/hip_runtime.h>
#include <hip/hip_bf16.h>
#include <math.h>

typedef __attribute__((ext_vector_type(2))) float v2f;
typedef __attribute__((ext_vector_type(8))) float v8f;

#define D_MODEL 512
#define HDIM    256
#define CAT_VOCAB 512
#define TIME_START 512
#define NOTE_START 1512
#define VEL_START 1640
#define DUR_START 1672

// ---------------- workspace layout ----------------
// [0]            : u32 counter of compacted vel tokens
// [256 ...]      : u32 list of vel token flat positions (up to n_tok)

__global__ void mhe_init_kernel(unsigned* __restrict__ cnt) { *cnt = 0u; }

// ---------------- pass 1: one wave32 per token, lane owns 16 dims ----------------

__device__ __forceinline__ void t2v_write(float tn,
                                          const float* __restrict__ lw,
                                          const float* __restrict__ lb,
                                          const float* __restrict__ pw,
                                          const float* __restrict__ pb,
                                          float* __restrict__ orow, int lane) {
  float4* ow = (float4*)orow;
  if (lane < 16) {                       // linear half: dims [0,256)
    const int d0 = lane * 16;
#pragma unroll
    for (int q = 0; q < 4; ++q) {
      float4 v;
      v.x = fmaf(tn, lw[d0 + q*4 + 0], lb[d0 + q*4 + 0]);
      v.y = fmaf(tn, lw[d0 + q*4 + 1], lb[d0 + q*4 + 1]);
      v.z = fmaf(tn, lw[d0 + q*4 + 2], lb[d0 + q*4 + 2]);
      v.w = fmaf(tn, lw[d0 + q*4 + 3], lb[d0 + q*4 + 3]);
      ow[q] = v;
    }
  } else {                               // periodic half: dims [256,512)
    const int d0 = lane * 16 - 256;
#pragma unroll
    for (int q = 0; q < 4; ++q) {
      float4 v;
      v.x = __sinf(fmaf(tn, pw[d0 + q*4 + 0], pb[d0 + q*4 + 0]));
      v.y = __sinf(fmaf(tn, pw[d0 + q*4 + 1], pb[d0 + q*4 + 1]));
      v.z = __sinf(fmaf(tn, pw[d0 + q*4 + 2], pb[d0 + q*4 + 2]));
      v.w = __sinf(fmaf(tn, pw[d0 + q*4 + 3], pb[d0 + q*4 + 3]));
      ow[q] = v;
    }
  }
}

__device__ __forceinline__ void pitch_write(int p, float* __restrict__ orow, int lane) {
  const float pf  = (float)p;
  const float oct = floorf(pf * (1.0f / 12.0f));
  const float sem = pf - 12.0f * oct;
  const int g0 = lane * 4;               // 4 groups of 4 per lane
  float4* ow = (float4*)orow;
#pragma unroll
  for (int q = 0; q < 4; ++q) {
    const float fi = (float)(g0 + q);
    const float po = oct * exp2f(-fi);                      // octave / 2^i
    const float ps = sem * 0.5235987755982988f * (fi + 1.0f); // sem * pi/6 * (i+1)
    float4 v;
    v.x = __sinf(po); v.y = __cosf(po);
    v.z = __sinf(ps); v.w = __cosf(ps);
    ow[q] = v;
  }
}

__global__ void mhe_main_kernel(const int* __restrict__ tokens,
                                const float* __restrict__ token_emb,
                                const float* __restrict__ time_lin_w,
                                const float* __restrict__ time_lin_b,
                                const float* __restrict__ time_w,
                                const float* __restrict__ time_b,
                                const float* __restrict__ dur_lin_w,
                                const float* __restrict__ dur_lin_b,
                                const float* __restrict__ dur_w,
                                const float* __restrict__ dur_b,
                                float* __restrict__ out,
                                unsigned* __restrict__ vel_cnt,
                                unsigned* __restrict__ vel_list,
                                int n_tok) {
  const int lane = threadIdx.x & 31;
  const int t = blockIdx.x * (blockDim.x >> 5) + (threadIdx.x >> 5);
  if (t >= n_tok) return;
  const int tok = tokens[t];
  float* orow = out + (size_t)t * D_MODEL + lane * 16;

  if (tok < CAT_VOCAB) {
    const float4* er = (const float4*)(token_emb + (size_t)tok * D_MODEL + lane * 16);
    float4* ow = (float4*)orow;
#pragma unroll
    for (int q = 0; q < 4; ++q) ow[q] = er[q];
  } else if (tok < NOTE_START) {
    t2v_write((float)(tok - TIME_START) * 1e-3f,
              time_lin_w, time_lin_b, time_w, time_b, orow, lane);
  } else if (tok < VEL_START) {
    pitch_write(tok - NOTE_START, orow, lane);
  } else if (tok < DUR_START) {
    // compact vel token; its output row is produced by the WMMA pass
    if (lane == 0) {
      unsigned s = atomicAdd(vel_cnt, 1u);
      vel_list[s] = (unsigned)t;
    }
  } else {
    t2v_write((float)(tok - DUR_START) * 1e-3f,
              dur_lin_w, dur_lin_b, dur_w, dur_b, orow, lane);
  }
}

// ---------------- pass 2: WMMA f32 GEMM + LayerNorm on compacted vel tokens ----------------
// Tile: 16 tokens. h[16][256] in LDS (A), w2^T streamed from global (B),
// e[16][512] accumulated in 16x16 f32 WMMA tiles, then bias + LN + scatter.

__global__ void mhe_vel_kernel(const int* __restrict__ tokens,
                               const float* __restrict__ w1,
                               const float* __restrict__ b1,
                               const float* __restrict__ w2,   // [512][256] row-major
                               const float* __restrict__ b2,
                               const float* __restrict__ ln_g,
                               const float* __restrict__ ln_b,
                               const unsigned* __restrict__ vel_cnt,
                               const unsigned* __restrict__ vel_list,
                               float* __restrict__ out) {
  __shared__ float    hbuf[16][HDIM];
  __shared__ float    ebuf[16][D_MODEL + 16];
  __shared__ unsigned posbuf[16];

  const unsigned count = *vel_cnt;
  const unsigned base  = blockIdx.x * 16u;
  if (base >= count) return;                    // uniform early-exit

  const int tid = threadIdx.x;
  if (tid < 16)
    posbuf[tid] = (base + (unsigned)tid < count) ? vel_list[base + tid] : 0xFFFFFFFFu;
  __syncthreads();

  // h = gelu(vn * w1 + b1), 16 values per thread (16 tokens x 256)
  {
    const int m  = tid >> 4;
    const int k0 = (tid & 15) * 16;
    const unsigned pos = posbuf[m];
    if (pos != 0xFFFFFFFFu) {
      const float vn = (float)(tokens[pos] - VEL_START) * (1.0f / 32.0f);
#pragma unroll
      for (int j = 0; j < 16; ++j) {
        const int k = k0 + j;
        const float x = fmaf(vn, w1[k], b1[k]);
        hbuf[m][k] = 0.5f * x * (1.0f + erff(x * 0.70710678118654752f));
      }
    } else {
#pragma unroll
      for (int j = 0; j < 16; ++j) hbuf[m][k0 + j] = 0.0f;
    }
  }
  __syncthreads();

  const int wv   = tid >> 5;        // 8 waves
  const int lane = tid & 31;
  const int m16  = lane & 15;
  const int khalf = (lane >> 4) << 1;   // 0 for lanes 0-15, 2 for lanes 16-31
  const int mhi   = (lane >> 4) * 8;    // C/D row offset for high half-wave

  // each wave produces 4 of the 32 N-tiles (16 cols each)
#pragma unroll
  for (int tile = 0; tile < 4; ++tile) {
    const int nt = wv + tile * 8;
    const int n  = nt * 16 + m16;
    const float* __restrict__ w2n  = w2 + (size_t)n * HDIM + khalf;  // B = w2^T
    const float* __restrict__ hrow = &hbuf[m16][khalf];              // A
    v8f acc = {};
    for (int k0 = 0; k0 < HDIM; k0 += 4) {
      v2f a, b;
      a.x = hrow[k0]; a.y = hrow[k0 + 1];   // A 16x4 f32 frag (2 VGPRs/lane)
      b.x = w2n[k0];  b.y = w2n[k0 + 1];    // B 4x16 f32 frag
      acc = __builtin_amdgcn_wmma_f32_16x16x4_f32(
          false, a, false, b, (short)0, acc, false, false);
    }
    // spill 16x16 f32 D tile: VGPR j -> row (j + mhi), col = lane%16
#pragma unroll
    for (int j = 0; j < 8; ++j)
      ebuf[j + mhi][nt * 16 + m16] = acc[j];
  }
  __syncthreads();

  // LayerNorm + scatter: each wave owns 2 tokens
#pragma unroll
  for (int mm = 0; mm < 2; ++mm) {
    const int m = wv * 2 + mm;
    const unsigned pos = posbuf[m];
    float s = 0.0f, s2 = 0.0f;
#pragma unroll
    for (int j = 0; j < 16; ++j) {
      const int d = lane + j * 32;
      const float e = ebuf[m][d] + b2[d];
      s += e; s2 += e * e;
    }
#pragma unroll
    for (int off = 16; off >= 1; off >>= 1) {
      s  += __shfl_xor(s,  off, 32);
      s2 += __shfl_xor(s2, off, 32);
    }
    const float mu  = s * (1.0f / 512.0f);
    const float var = fmaf(s2, 1.0f / 512.0f, -mu * mu);
    const float inv = rsqrtf(var + 1e-5f);
    if (pos != 0xFFFFFFFFu) {
      float* orow = out + (size_t)pos * D_MODEL;
#pragma unroll
      for (int q = 0; q < 4; ++q) {
        const int d0 = lane * 16 + q * 4;
        float4 v;
        v.x = fmaf(ln_g[d0+0], (ebuf[m][d0+0] + b2[d0+0] - mu) * inv, ln_b[d0+0]);
        v.y = fmaf(ln_g[d0+1], (ebuf[m][d0+1] + b2[d0+1] - mu) * inv, ln_b[d0+1]);
        v.z = fmaf(ln_g[d0+2], (ebuf[m][d0+2] + b2[d0+2] - mu) * inv, ln_b[d0+2]);
        v.w = fmaf(ln_g[d0+3], (ebuf[m][d0+3] + b2[d0+3] - mu) * inv, ln_b[d0+3]);
        *(float4*)(orow + d0) = v;
      }
    }
  }
}

extern "C" void kernel_launch(void* const* d_in, const int* in_sizes, int n_in,
                              void* d_out, int out_size, void* d_ws, size_t ws_size,
                              hipStream_t stream) {
  const int*   tokens     = (const int*)  d_in[0];
  const float* token_emb  = (const float*)d_in[1];
  const float* time_lin_w = (const float*)d_in[2];
  const float* time_lin_b = (const float*)d_in[3];
  const float* time_w     = (const float*)d_in[4];
  const float* time_b     = (const float*)d_in[5];
  const float* vel_w1     = (const float*)d_in[6];
  const float* vel_b1     = (const float*)d_in[7];
  const float* vel_w2     = (const float*)d_in[8];
  const float* vel_b2     = (const float*)d_in[9];
  const float* vel_ln_g   = (const float*)d_in[10];
  const float* vel_ln_b   = (const float*)d_in[11];
  const float* dur_lin_w  = (const float*)d_in[12];
  const float* dur_lin_b  = (const float*)d_in[13];
  const float* dur_w      = (const float*)d_in[14];
  const float* dur_b      = (const float*)d_in[15];
  float* out = (float*)d_out;

  const int n_tok = in_sizes[0];                 // 8 * 8192 = 65536

  unsigned* vel_cnt  = (unsigned*)d_ws;
  unsigned* vel_list = (unsigned*)((char*)d_ws + 256);

  mhe_init_kernel<<<1, 1, 0, stream>>>(vel_cnt);

  const int blocks1 = (n_tok + 7) / 8;           // 8 waves/block, wave per token
  mhe_main_kernel<<<blocks1, 256, 0, stream>>>(
      tokens, token_emb, time_lin_w, time_lin_b, time_w, time_b,
      dur_lin_w, dur_lin_b, dur_w, dur_b, out, vel_cnt, vel_list, n_tok);

  const int blocks2 = (n_tok + 15) / 16;         // worst case; blocks past count exit
  mhe_vel_kernel<<<blocks2, 256, 0, stream>>>(
      tokens, vel_w1, vel_b1, vel_w2, vel_b2, vel_ln_g, vel_ln_b,
      vel_cnt, vel_list, out);
}